// BinSAGE_67568425500673
// MI455X (gfx1250) — compile-verified
//
#include <hip/hip_runtime.h>
#include <cstddef>
#include <cstdint>

// ---------------- problem constants (from reference) ----------------
#define IN_C  100
#define HID   256
#define OUT_C 47
#define N1C   100000
#define N2C   25000
#define N3C   4096
#define E0C   1500000
#define E1C   250000
#define E2C   40960

typedef __attribute__((ext_vector_type(2))) float v2f;
typedef __attribute__((ext_vector_type(8))) float v8f;

static inline int ceil_div(long long a, long long b) { return (int)((a + b - 1) / b); }

// ---------------- utility: zero a float buffer ----------------
__global__ void zero_f32(float* __restrict__ p, size_t n) {
    size_t i = (size_t)blockIdx.x * blockDim.x + threadIdx.x;
    if (i < n) p[i] = 0.0f;
}

// ---------------- weight transpose: WT[n,k] = W[k,n] ----------------
// Weights are tiny (<= 256x256); done once per launch so the GEMM's B
// operand becomes K-contiguous per lane (b64 loads, immediate offsets).
__global__ void transpose_w(const float* __restrict__ W, float* __restrict__ WT,
                            int K, int N) {
    int i = blockIdx.x * blockDim.x + threadIdx.x;
    if (i >= K * N) return;
    int k = i / N;
    int n = i - k * N;
    WT[(size_t)n * K + k] = W[i];
}

// ---------------- edge-parallel gather + atomic scatter-add ----------------
// One thread per (edge, 4-float chunk). F must be a multiple of 4
// (100 -> 25 chunks, 256 -> 64 chunks). Chunk 0 also bumps the count.
// Lowers to GLOBAL_ATOMIC_ADD_F32 (no-return -> STOREcnt path).
__global__ void scatter_mean_accum(const float* __restrict__ x,
                                   const int*  __restrict__ src,
                                   const int*  __restrict__ dst,
                                   float* __restrict__ agg,
                                   float* __restrict__ cnt,
                                   int E, int F) {
    const int chunks = F >> 2;
    long long idx = (long long)blockIdx.x * blockDim.x + threadIdx.x;
    long long total = (long long)E * chunks;
    if (idx >= total) return;
    int e = (int)(idx / chunks);
    int c = (int)(idx - (long long)e * chunks);
    int s = src[e];
    int d = dst[e];
    const float4 v = *(const float4*)(x + (size_t)s * F + (size_t)c * 4);
    float* ap = agg + (size_t)d * F + (size_t)c * 4;
    atomicAdd(ap + 0, v.x);
    atomicAdd(ap + 1, v.y);
    atomicAdd(ap + 2, v.z);
    atomicAdd(ap + 3, v.w);
    if (c == 0) atomicAdd(cnt + d, 1.0f);
}

// ---------------- agg /= max(cnt, 1) ----------------
__global__ void mean_div(float* __restrict__ agg, const float* __restrict__ cnt,
                         int n, int F) {
    size_t i = (size_t)blockIdx.x * blockDim.x + threadIdx.x;
    if (i >= (size_t)n * F) return;
    int row = (int)(i / (size_t)F);
    agg[i] = agg[i] / fmaxf(cnt[row], 1.0f);
}

// ---------------- fused SAGE GEMM: out = agg@Wl + xt@Wr + b ----------------
// Wave computes a 16x32 output tile with V_WMMA_F32_16X16X4_F32 (exact fp32):
// one A fragment feeds two independent accumulator chains (cols 0..15, 16..31).
// Weights are pre-transposed to [N,K], so BOTH operands are K-contiguous per
// lane: each fragment is one global_load_b64 at a running immediate offset.
// Out-of-range rows/cols are CLAMPED on load (their products only land in
// rows/cols that are masked at store), so the inner loop has full EXEC and
// no branches: pure clause-able loads + WMMA.
//   A frag (16x4): lane l -> row tm*16+(l&15), K = k0 + 2*(l>>4) (+1 in .y)
//   B frag (4x16): lane l -> col        (l&15), same K split
//   D: vgpr r, lane l -> m = r + 8*(l>>4), n = l&15
__global__ void sage_gemm_wmma(const float* __restrict__ agg,   // [M,K]
                               const float* __restrict__ xt,    // [M,K]
                               const float* __restrict__ WlT,   // [N,K] (transposed)
                               const float* __restrict__ WrT,   // [N,K] (transposed)
                               const float* __restrict__ bias,  // [N]
                               float* __restrict__ out,         // [M,N]
                               int M, int N, int K) {
    const int lane = threadIdx.x & 31;
    const int wave = threadIdx.x >> 5;
    const int wavesPerBlock = blockDim.x >> 5;

    const int tilesM = (M + 15) >> 4;
    const int tilesN = (N + 31) >> 5;          // 32 columns per wave
    const int tile = blockIdx.x * wavesPerBlock + wave;
    if (tile >= tilesM * tilesN) return;       // wave-uniform: EXEC stays all-1s

    const int tm = tile / tilesN;
    const int tn = tile - tm * tilesN;

    const int halfSel = lane >> 4;             // 0 -> K {k0,k0+1}; 1 -> {k0+2,k0+3}
    const int l15     = lane & 15;

    const int rowReal  = tm * 16 + l15;
    const int colReal0 = tn * 32 + l15;
    const int colReal1 = colReal0 + 16;
    const int row  = (rowReal  < M) ? rowReal  : (M - 1);   // clamp, mask at store
    const int col0 = (colReal0 < N) ? colReal0 : (N - 1);
    const int col1 = (colReal1 < N) ? colReal1 : (N - 1);

    v8f acc0 = {};
    v8f acc1 = {};

    #pragma unroll
    for (int pass = 0; pass < 2; ++pass) {
        const float* __restrict__ A  = pass ? xt  : agg;
        const float* __restrict__ WT = pass ? WrT : WlT;
        // all three streams: K-contiguous, running pointers, no inner multiplies
        const float* ap  = A  + (size_t)row  * K + (halfSel << 1);
        const float* wp0 = WT + (size_t)col0 * K + (halfSel << 1);
        const float* wp1 = WT + (size_t)col1 * K + (halfSel << 1);
        for (int k0 = 0; k0 < K; k0 += 4) {
            const v2f a  = *(const v2f*)ap;    // global_load_b64
            const v2f b0 = *(const v2f*)wp0;   // global_load_b64
            const v2f b1 = *(const v2f*)wp1;   // global_load_b64
            acc0 = __builtin_amdgcn_wmma_f32_16x16x4_f32(
                       false, a, false, b0, (short)0, acc0, false, false);
            acc1 = __builtin_amdgcn_wmma_f32_16x16x4_f32(
                       false, a, false, b1, (short)0, acc1, false, false);
            ap  += 4;
            wp0 += 4;
            wp1 += 4;
        }
    }

    const float bv0 = bias[col0];              // clamped-safe; masked below
    const float bv1 = bias[col1];
    const bool ok0 = (colReal0 < N);
    const bool ok1 = (colReal1 < N);
    #pragma unroll
    for (int r = 0; r < 8; ++r) {
        const int m = tm * 16 + (halfSel << 3) + r;
        if (m < M) {
            if (ok0) out[(size_t)m * N + colReal0] = acc0[r] + bv0;
            if (ok1) out[(size_t)m * N + colReal1] = acc1[r] + bv1;
        }
    }
}

// ---------------- wave-per-row log_softmax (in place) ----------------
__global__ void log_softmax_rows(float* __restrict__ p, int M, int N) {
    const int lane = threadIdx.x & 31;
    const int row  = blockIdx.x * (blockDim.x >> 5) + (threadIdx.x >> 5);
    if (row >= M) return;
    float* r = p + (size_t)row * N;

    float mx = -3.402823466e+38f;
    for (int c = lane; c < N; c += 32) mx = fmaxf(mx, r[c]);
    #pragma unroll
    for (int off = 16; off; off >>= 1) mx = fmaxf(mx, __shfl_xor(mx, off, 32));

    float sum = 0.0f;
    for (int c = lane; c < N; c += 32) sum += expf(r[c] - mx);
    #pragma unroll
    for (int off = 16; off; off >>= 1) sum += __shfl_xor(sum, off, 32);

    const float lse = mx + logf(sum);
    for (int c = lane; c < N; c += 32) r[c] = r[c] - lse;
}

// ---------------- host-side orchestration ----------------
extern "C" void kernel_launch(void* const* d_in, const int* in_sizes, int n_in,
                              void* d_out, int out_size, void* d_ws, size_t ws_size,
                              hipStream_t stream) {
    (void)in_sizes; (void)n_in; (void)out_size; (void)ws_size;

    const float* x    = (const float*)d_in[0];
    const int*   src0 = (const int*)d_in[1];
    const int*   dst0 = (const int*)d_in[2];
    const int*   src1 = (const int*)d_in[3];
    const int*   dst1 = (const int*)d_in[4];
    const int*   src2 = (const int*)d_in[5];
    const int*   dst2 = (const int*)d_in[6];
    // d_in[7..9] = n1,n2,n3 scalars (known constants)
    const float* Wl0 = (const float*)d_in[10];
    const float* Wr0 = (const float*)d_in[11];
    const float* b0  = (const float*)d_in[12];
    const float* Wl1 = (const float*)d_in[13];
    const float* Wr1 = (const float*)d_in[14];
    const float* b1  = (const float*)d_in[15];
    const float* Wl2 = (const float*)d_in[16];
    const float* Wr2 = (const float*)d_in[17];
    const float* b2  = (const float*)d_in[18];

    float* logits = (float*)d_out;   // [N3C, OUT_C]

    // Workspace carving with lifetime-based reuse:
    //  region A: agg+cnt for all three layers (max footprint = layer 0)
    //  region B: h1 [N1C, HID]
    //  region C: h2 [N2C, HID]
    //  region D: six transposed weight matrices (~0.8 MB)
    const size_t szAgg0 = (size_t)N1C * IN_C + N1C;       // 10,100,000 f32
    float* regA = (float*)d_ws;
    float* h1   = regA + szAgg0;                          // 25,600,000 f32
    float* h2   = h1 + (size_t)N1C * HID;                 //  6,400,000 f32
    float* WlT0 = h2 + (size_t)N2C * HID;
    float* WrT0 = WlT0 + (size_t)IN_C * HID;
    float* WlT1 = WrT0 + (size_t)IN_C * HID;
    float* WrT1 = WlT1 + (size_t)HID * HID;
    float* WlT2 = WrT1 + (size_t)HID * HID;
    float* WrT2 = WlT2 + (size_t)HID * OUT_C;

    const int T = 256;

    // ---------------- transpose all weights up front ----------------
    transpose_w<<<ceil_div((long long)IN_C * HID, T), T, 0, stream>>>(Wl0, WlT0, IN_C, HID);
    transpose_w<<<ceil_div((long long)IN_C * HID, T), T, 0, stream>>>(Wr0, WrT0, IN_C, HID);
    transpose_w<<<ceil_div((long long)HID * HID,  T), T, 0, stream>>>(Wl1, WlT1, HID, HID);
    transpose_w<<<ceil_div((long long)HID * HID,  T), T, 0, stream>>>(Wr1, WrT1, HID, HID);
    transpose_w<<<ceil_div((long long)HID * OUT_C,T), T, 0, stream>>>(Wl2, WlT2, HID, OUT_C);
    transpose_w<<<ceil_div((long long)HID * OUT_C,T), T, 0, stream>>>(Wr2, WrT2, HID, OUT_C);

    // ---------------- layer 0: (x, src0, dst0) -> h1 [N1C, HID] ----------------
    {
        float* agg = regA;
        float* cnt = regA + (size_t)N1C * IN_C;
        size_t zn = (size_t)N1C * IN_C + N1C;
        zero_f32<<<ceil_div((long long)zn, T), T, 0, stream>>>(regA, zn);

        long long work = (long long)E0C * (IN_C / 4);
        scatter_mean_accum<<<ceil_div(work, T), T, 0, stream>>>(
            x, src0, dst0, agg, cnt, E0C, IN_C);

        mean_div<<<ceil_div((long long)N1C * IN_C, T), T, 0, stream>>>(
            agg, cnt, N1C, IN_C);

        const int tiles = ((N1C + 15) / 16) * ((HID + 31) / 32);
        sage_gemm_wmma<<<ceil_div(tiles, T / 32), T, 0, stream>>>(
            agg, x, WlT0, WrT0, b0, h1, N1C, HID, IN_C);
    }

    // ---------------- layer 1: (h1, src1, dst1) -> h2 [N2C, HID] ----------------
    {
        float* agg = regA;
        float* cnt = regA + (size_t)N2C * HID;
        size_t zn = (size_t)N2C * HID + N2C;
        zero_f32<<<ceil_div((long long)zn, T), T, 0, stream>>>(regA, zn);

        long long work = (long long)E1C * (HID / 4);
        scatter_mean_accum<<<ceil_div(work, T), T, 0, stream>>>(
            h1, src1, dst1, agg, cnt, E1C, HID);

        mean_div<<<ceil_div((long long)N2C * HID, T), T, 0, stream>>>(
            agg, cnt, N2C, HID);

        const int tiles = ((N2C + 15) / 16) * ((HID + 31) / 32);
        sage_gemm_wmma<<<ceil_div(tiles, T / 32), T, 0, stream>>>(
            agg, h1, WlT1, WrT1, b1, h2, N2C, HID, HID);
    }

    // ---------------- layer 2: (h2, src2, dst2) -> logits [N3C, OUT_C] ----------
    {
        float* agg = regA;
        float* cnt = regA + (size_t)N3C * HID;
        size_t zn = (size_t)N3C * HID + N3C;
        zero_f32<<<ceil_div((long long)zn, T), T, 0, stream>>>(regA, zn);

        long long work = (long long)E2C * (HID / 4);
        scatter_mean_accum<<<ceil_div(work, T), T, 0, stream>>>(
            h2, src2, dst2, agg, cnt, E2C, HID);

        mean_div<<<ceil_div((long long)N3C * HID, T), T, 0, stream>>>(
            agg, cnt, N3C, HID);

        const int tiles = ((N3C + 15) / 16) * ((OUT_C + 31) / 32);
        sage_gemm_wmma<<<ceil_div(tiles, T / 32), T, 0, stream>>>(
            agg, h2, WlT2, WrT2, b2, logits, N3C, OUT_C, HID);
    }

    // ---------------- log_softmax over rows of d_out ----------------
    log_softmax_rows<<<ceil_div(N3C, T / 32), T, 0, stream>>>(logits, N3C, OUT_C);
}